// MPNN_62088047231705
// MI455X (gfx1250) — compile-verified
//
#include <hip/hip_runtime.h>
#include <math.h>

typedef __attribute__((ext_vector_type(2))) float v2f;
typedef __attribute__((ext_vector_type(8))) float v8f;

#define HD  128   // hidden width
#define TILE 32   // rows (edges/nodes) per workgroup
#define STS 516   // LDS state stride in floats (pad: conflict-free ds_load_b64)
#define HS  132   // LDS hidden stride in floats

__device__ inline v8f vzero() {
  v8f z = {0.f, 0.f, 0.f, 0.f, 0.f, 0.f, 0.f, 0.f};
  return z;
}

__device__ inline float siluf(float x) { return x / (1.f + __expf(-x)); }

__device__ inline float tanhf_(float x) {
  float xc = fminf(fmaxf(x, -15.f), 15.f);
  float t = __expf(2.f * xc);
  return (t - 1.f) / (t + 1.f);
}

__device__ inline void fatomic_add(float* p, float v) {
  __hip_atomic_fetch_add(p, v, __ATOMIC_RELAXED, __HIP_MEMORY_SCOPE_AGENT);
}

// Accumulate a 32x16 output tile (two 16x16 WMMA row-tiles) of
//   state[rows, colBase .. colBase+len) @ W[wRowBase .. wRowBase+len, c0 .. c0+16)
// A operand from LDS (stride ldsStride floats), B operand streamed from global.
// f32 WMMA layout (wave32): A: lanes 0-15 hold K=k0,k0+1; lanes 16-31 K=k0+2,k0+3.
// B mirrors A; C/D: VGPR v holds row v (lanes 0-15) / v+8 (lanes 16-31), col = lane&15.
__device__ inline void mma_seg(v8f& a0, v8f& a1,
                               const float* lds, int ldsStride, int colBase,
                               const float* __restrict__ W, int wRowBase, int len,
                               int c0, int lane)
{
  const int m  = lane & 15;
  const int kh = lane >> 4;
  const int n  = lane & 15;
  const float* ap = lds + m * ldsStride + colBase + 2 * kh;
  const float* wp = W + (wRowBase + 2 * kh) * HD + c0 + n;
  const int rowOfs = 16 * ldsStride;
  #pragma unroll 4
  for (int k = 0; k < len; k += 4) {
    v2f B;  B.x  = wp[0];       B.y  = wp[HD];
    v2f A0; A0.x = ap[0];       A0.y = ap[1];
    v2f A1; A1.x = ap[rowOfs];  A1.y = ap[rowOfs + 1];
    a0 = __builtin_amdgcn_wmma_f32_16x16x4_f32(false, A0, false, B, (short)0, a0, false, false);
    a1 = __builtin_amdgcn_wmma_f32_16x16x4_f32(false, A1, false, B, (short)0, a1, false, false);
    ap += 4;
    wp += 4 * HD;
  }
}

// ---------------------------------------------------------------------------
// Embedding: h = [x, pe] @ We + be  (K=56); pe_h = pe @ Wpe + bpe (K=24)
// ---------------------------------------------------------------------------
#define EMS 60  // embed state stride
__global__ __launch_bounds__(256)
void embed_kernel(const float* __restrict__ x, const float* __restrict__ pe,
                  const float* __restrict__ eW, const float* __restrict__ eB,
                  const float* __restrict__ pW, const float* __restrict__ pB,
                  float* __restrict__ h, float* __restrict__ peh, int N)
{
  __shared__ float st[TILE * EMS];
  const int tid = threadIdx.x;
  const int nodeBase = blockIdx.x * TILE;
  const int vN = min(TILE, N - nodeBase);

  for (int i = tid; i < TILE * 56; i += 256) {
    int e = i / 56, c = i - e * 56;
    int node = min(nodeBase + e, N - 1);
    st[e * EMS + c] = (c < 32) ? x[node * 32 + c] : pe[node * 24 + (c - 32)];
  }
  __syncthreads();

  const int lane = tid & 31;
  const int c0 = (tid >> 5) * 16;
  const int n = lane & 15;
  const int hh = lane >> 4;

  v8f a0 = vzero(), a1 = vzero();
  mma_seg(a0, a1, st, EMS, 0, eW, 0, 56, c0, lane);
  {
    float b = eB[c0 + n];
    #pragma unroll
    for (int v = 0; v < 8; ++v) {
      int m0 = v + 8 * hh, m1 = m0 + 16;
      if (m0 < vN) h[(nodeBase + m0) * HD + c0 + n] = a0[v] + b;
      if (m1 < vN) h[(nodeBase + m1) * HD + c0 + n] = a1[v] + b;
    }
  }
  a0 = vzero(); a1 = vzero();
  mma_seg(a0, a1, st, EMS, 32, pW, 0, 24, c0, lane);
  {
    float b = pB[c0 + n];
    #pragma unroll
    for (int v = 0; v < 8; ++v) {
      int m0 = v + 8 * hh, m1 = m0 + 16;
      if (m0 < vN) peh[(nodeBase + m0) * HD + c0 + n] = a0[v] + b;
      if (m1 < vN) peh[(nodeBase + m1) * HD + c0 + n] = a1[v] + b;
    }
  }
}

// ---------------------------------------------------------------------------
// Edge kernel: state = [h[s], pe_h[s], h[r], pe_h[r], dist] (K=513)
//   message = silu(silu(state@mW1+mB1)@mW2+mB2)        -> atomic add aggr[r]
//   pos_msg = tanh(tanh(pe_state@pW1+pB1)@pW2+pB2)     -> atomic add posAggr[r]
// ---------------------------------------------------------------------------
__global__ __launch_bounds__(256)
void lspe_edge_kernel(const float* __restrict__ h, const float* __restrict__ peh,
                      const float* __restrict__ pos, const int* __restrict__ ei,
                      const float* __restrict__ mW1, const float* __restrict__ mB1,
                      const float* __restrict__ mW2, const float* __restrict__ mB2,
                      const float* __restrict__ pW1, const float* __restrict__ pB1,
                      const float* __restrict__ pW2, const float* __restrict__ pB2,
                      float* __restrict__ aggr, float* __restrict__ posAggr, int E)
{
  __shared__ float st[TILE * STS];
  __shared__ float hid[TILE * HS];
  __shared__ int sIdx[TILE], rIdx[TILE];

  const int tid = threadIdx.x;
  const int eBase = blockIdx.x * TILE;
  const int vE = min(TILE, E - eBase);

  if (tid < TILE) {
    int e = min(eBase + tid, E - 1);
    int s = ei[e];
    int r = ei[E + e];
    sIdx[tid] = s;
    rIdx[tid] = r;
    float dx = pos[3 * s + 0] - pos[3 * r + 0];
    float dy = pos[3 * s + 1] - pos[3 * r + 1];
    float dz = pos[3 * s + 2] - pos[3 * r + 2];
    st[tid * STS + 512] = sqrtf(dx * dx + dy * dy + dz * dz);
  }
  __syncthreads();

  // gather: seg0 h[s] | seg1 pe_h[s] | seg2 h[r] | seg3 pe_h[r] -> cols seg*128
  for (int i = tid; i < TILE * 128; i += 256) {
    int e = i >> 7;
    int seg = (i >> 5) & 3;
    int c4 = i & 31;
    int node = (seg < 2) ? sIdx[e] : rIdx[e];
    const float* src = ((seg & 1) ? peh : h) + node * HD + c4 * 4;
    float4 v = *(const float4*)src;
    *(float4*)&st[e * STS + (seg << 7) + c4 * 4] = v;
  }
  __syncthreads();

  const int lane = tid & 31;
  const int c0 = (tid >> 5) * 16;
  const int n = lane & 15;
  const int hh = lane >> 4;

  // ---- message MLP ----
  v8f a0 = vzero(), a1 = vzero();
  mma_seg(a0, a1, st, STS, 0, mW1, 0, 512, c0, lane);
  {
    float wd = mW1[512 * HD + c0 + n];  // dist row (rank-1 for K=513th col)
    float b = mB1[c0 + n];
    #pragma unroll
    for (int v = 0; v < 8; ++v) {
      int m0 = v + 8 * hh, m1 = m0 + 16;
      float x0 = a0[v] + st[m0 * STS + 512] * wd + b;
      float x1 = a1[v] + st[m1 * STS + 512] * wd + b;
      hid[m0 * HS + c0 + n] = siluf(x0);
      hid[m1 * HS + c0 + n] = siluf(x1);
    }
  }
  __syncthreads();
  a0 = vzero(); a1 = vzero();
  mma_seg(a0, a1, hid, HS, 0, mW2, 0, 128, c0, lane);
  {
    float b = mB2[c0 + n];
    #pragma unroll
    for (int v = 0; v < 8; ++v) {
      int m0 = v + 8 * hh, m1 = m0 + 16;
      float x0 = siluf(a0[v] + b);
      float x1 = siluf(a1[v] + b);
      if (m0 < vE) fatomic_add(&aggr[rIdx[m0] * HD + c0 + n], x0);
      if (m1 < vE) fatomic_add(&aggr[rIdx[m1] * HD + c0 + n], x1);
    }
  }
  __syncthreads();

  // ---- positional MLP (pe_state = cols 128..255 and 384..511 + dist) ----
  a0 = vzero(); a1 = vzero();
  mma_seg(a0, a1, st, STS, 128, pW1, 0, 128, c0, lane);
  mma_seg(a0, a1, st, STS, 384, pW1, 128, 128, c0, lane);
  {
    float wd = pW1[256 * HD + c0 + n];
    float b = pB1[c0 + n];
    #pragma unroll
    for (int v = 0; v < 8; ++v) {
      int m0 = v + 8 * hh, m1 = m0 + 16;
      float x0 = a0[v] + st[m0 * STS + 512] * wd + b;
      float x1 = a1[v] + st[m1 * STS + 512] * wd + b;
      hid[m0 * HS + c0 + n] = tanhf_(x0);
      hid[m1 * HS + c0 + n] = tanhf_(x1);
    }
  }
  __syncthreads();
  a0 = vzero(); a1 = vzero();
  mma_seg(a0, a1, hid, HS, 0, pW2, 0, 128, c0, lane);
  {
    float b = pB2[c0 + n];
    #pragma unroll
    for (int v = 0; v < 8; ++v) {
      int m0 = v + 8 * hh, m1 = m0 + 16;
      float x0 = tanhf_(a0[v] + b);
      float x1 = tanhf_(a1[v] + b);
      if (m0 < vE) fatomic_add(&posAggr[rIdx[m0] * HD + c0 + n], x0);
      if (m1 < vE) fatomic_add(&posAggr[rIdx[m1] * HD + c0 + n], x1);
    }
  }
}

// ---------------------------------------------------------------------------
// Node kernel: h += silu([h,pe_h,aggr]@uW1+uB1)@uW2+uB2
//              pe_h += tanh(tanh([pe_h,posAggr]@qW1+qB1)@qW2+qB2)
// ---------------------------------------------------------------------------
__global__ __launch_bounds__(256)
void lspe_node_kernel(float* __restrict__ h, float* __restrict__ peh,
                      const float* __restrict__ aggr, const float* __restrict__ posAggr,
                      const float* __restrict__ uW1, const float* __restrict__ uB1,
                      const float* __restrict__ uW2, const float* __restrict__ uB2,
                      const float* __restrict__ qW1, const float* __restrict__ qB1,
                      const float* __restrict__ qW2, const float* __restrict__ qB2,
                      int N)
{
  __shared__ float st[TILE * STS];
  __shared__ float hid[TILE * HS];
  const int tid = threadIdx.x;
  const int nodeBase = blockIdx.x * TILE;
  const int vN = min(TILE, N - nodeBase);

  // cols: 0..127 h | 128..255 pe_h | 256..383 aggr | 384..511 posAggr
  for (int i = tid; i < TILE * 128; i += 256) {
    int e = i >> 7;
    int seg = (i >> 5) & 3;
    int c4 = i & 31;
    int node = min(nodeBase + e, N - 1);
    const float* base = (seg == 0) ? h : (seg == 1) ? peh : (seg == 2) ? aggr : posAggr;
    float4 v = *(const float4*)(base + node * HD + c4 * 4);
    *(float4*)&st[e * STS + (seg << 7) + c4 * 4] = v;
  }
  __syncthreads();

  const int lane = tid & 31;
  const int c0 = (tid >> 5) * 16;
  const int n = lane & 15;
  const int hh = lane >> 4;

  // ---- upd: K=384 contiguous, silu then linear ----
  v8f a0 = vzero(), a1 = vzero();
  mma_seg(a0, a1, st, STS, 0, uW1, 0, 384, c0, lane);
  {
    float b = uB1[c0 + n];
    #pragma unroll
    for (int v = 0; v < 8; ++v) {
      int m0 = v + 8 * hh, m1 = m0 + 16;
      hid[m0 * HS + c0 + n] = siluf(a0[v] + b);
      hid[m1 * HS + c0 + n] = siluf(a1[v] + b);
    }
  }
  __syncthreads();
  a0 = vzero(); a1 = vzero();
  mma_seg(a0, a1, hid, HS, 0, uW2, 0, 128, c0, lane);
  {
    float b = uB2[c0 + n];
    #pragma unroll
    for (int v = 0; v < 8; ++v) {
      int m0 = v + 8 * hh, m1 = m0 + 16;
      if (m0 < vN) { float* p = &h[(nodeBase + m0) * HD + c0 + n]; *p += a0[v] + b; }
      if (m1 < vN) { float* p = &h[(nodeBase + m1) * HD + c0 + n]; *p += a1[v] + b; }
    }
  }
  __syncthreads();

  // ---- upd_pe: [pe_h, posAggr] = cols 128..255 and 384..511, tanh/tanh ----
  a0 = vzero(); a1 = vzero();
  mma_seg(a0, a1, st, STS, 128, qW1, 0, 128, c0, lane);
  mma_seg(a0, a1, st, STS, 384, qW1, 128, 128, c0, lane);
  {
    float b = qB1[c0 + n];
    #pragma unroll
    for (int v = 0; v < 8; ++v) {
      int m0 = v + 8 * hh, m1 = m0 + 16;
      hid[m0 * HS + c0 + n] = tanhf_(a0[v] + b);
      hid[m1 * HS + c0 + n] = tanhf_(a1[v] + b);
    }
  }
  __syncthreads();
  a0 = vzero(); a1 = vzero();
  mma_seg(a0, a1, hid, HS, 0, qW2, 0, 128, c0, lane);
  {
    float b = qB2[c0 + n];
    #pragma unroll
    for (int v = 0; v < 8; ++v) {
      int m0 = v + 8 * hh, m1 = m0 + 16;
      if (m0 < vN) { float* p = &peh[(nodeBase + m0) * HD + c0 + n]; *p += tanhf_(a0[v] + b); }
      if (m1 < vN) { float* p = &peh[(nodeBase + m1) * HD + c0 + n]; *p += tanhf_(a1[v] + b); }
    }
  }
}

// ---------------------------------------------------------------------------
// pre_readout + global_add_pool: pooled[batch] += (silu(h@W1+b1))@W2+b2
// ---------------------------------------------------------------------------
__global__ __launch_bounds__(256)
void prepool_kernel(const float* __restrict__ h, const int* __restrict__ batch,
                    const float* __restrict__ W1, const float* __restrict__ B1,
                    const float* __restrict__ W2, const float* __restrict__ B2,
                    float* __restrict__ pooled, int N)
{
  __shared__ float st[TILE * HS];
  __shared__ float hid[TILE * HS];
  __shared__ int bidx[TILE];
  const int tid = threadIdx.x;
  const int nodeBase = blockIdx.x * TILE;
  const int vN = min(TILE, N - nodeBase);

  if (tid < TILE) {
    int node = min(nodeBase + tid, N - 1);
    bidx[tid] = batch[node];
  }
  for (int i = tid; i < TILE * 32; i += 256) {
    int e = i >> 5, c4 = i & 31;
    int node = min(nodeBase + e, N - 1);
    *(float4*)&st[e * HS + c4 * 4] = *(const float4*)&h[node * HD + c4 * 4];
  }
  __syncthreads();

  const int lane = tid & 31;
  const int c0 = (tid >> 5) * 16;
  const int n = lane & 15;
  const int hh = lane >> 4;

  v8f a0 = vzero(), a1 = vzero();
  mma_seg(a0, a1, st, HS, 0, W1, 0, 128, c0, lane);
  {
    float b = B1[c0 + n];
    #pragma unroll
    for (int v = 0; v < 8; ++v) {
      int m0 = v + 8 * hh, m1 = m0 + 16;
      hid[m0 * HS + c0 + n] = siluf(a0[v] + b);
      hid[m1 * HS + c0 + n] = siluf(a1[v] + b);
    }
  }
  __syncthreads();
  a0 = vzero(); a1 = vzero();
  mma_seg(a0, a1, hid, HS, 0, W2, 0, 128, c0, lane);
  {
    float b = B2[c0 + n];
    #pragma unroll
    for (int v = 0; v < 8; ++v) {
      int m0 = v + 8 * hh, m1 = m0 + 16;
      if (m0 < vN) fatomic_add(&pooled[bidx[m0] * HD + c0 + n], a0[v] + b);
      if (m1 < vN) fatomic_add(&pooled[bidx[m1] * HD + c0 + n], a1[v] + b);
    }
  }
}

// ---------------------------------------------------------------------------
// Readout: out[g] = silu(pooled@W1+b1) . w2 + b2
// ---------------------------------------------------------------------------
__global__ __launch_bounds__(256)
void readout_kernel(const float* __restrict__ pooled,
                    const float* __restrict__ W1, const float* __restrict__ B1,
                    const float* __restrict__ w2, const float* __restrict__ b2,
                    float* __restrict__ out, int G)
{
  __shared__ float st[TILE * HS];
  __shared__ float hid[TILE * HS];
  __shared__ float red[TILE * 8];
  const int tid = threadIdx.x;
  const int gBase = blockIdx.x * TILE;

  for (int i = tid; i < TILE * 32; i += 256) {
    int e = i >> 5, c4 = i & 31;
    int g = min(gBase + e, G - 1);
    *(float4*)&st[e * HS + c4 * 4] = *(const float4*)&pooled[g * HD + c4 * 4];
  }
  __syncthreads();

  const int lane = tid & 31;
  const int c0 = (tid >> 5) * 16;
  const int n = lane & 15;
  const int hh = lane >> 4;

  v8f a0 = vzero(), a1 = vzero();
  mma_seg(a0, a1, st, HS, 0, W1, 0, 128, c0, lane);
  {
    float b = B1[c0 + n];
    #pragma unroll
    for (int v = 0; v < 8; ++v) {
      int m0 = v + 8 * hh, m1 = m0 + 16;
      hid[m0 * HS + c0 + n] = siluf(a0[v] + b);
      hid[m1 * HS + c0 + n] = siluf(a1[v] + b);
    }
  }
  __syncthreads();

  int r = tid >> 3, p = tid & 7;
  float s = 0.f;
  #pragma unroll
  for (int k = 0; k < 16; ++k) s += hid[r * HS + p * 16 + k] * w2[p * 16 + k];
  red[r * 8 + p] = s;
  __syncthreads();
  if (p == 0) {
    float t = b2[0];
    #pragma unroll
    for (int j = 0; j < 8; ++j) t += red[r * 8 + j];
    if (gBase + r < G) out[gBase + r] = t;
  }
}

// ---------------------------------------------------------------------------
extern "C" void kernel_launch(void* const* d_in, const int* in_sizes, int n_in,
                              void* d_out, int out_size, void* d_ws, size_t ws_size,
                              hipStream_t stream) {
  (void)n_in; (void)ws_size;
  const float* x   = (const float*)d_in[0];
  const float* pos = (const float*)d_in[1];
  const float* pe  = (const float*)d_in[2];
  const int* ei    = (const int*)d_in[3];
  const int* batch = (const int*)d_in[4];
  auto F = [&](int i) { return (const float*)d_in[i]; };

  const int N = in_sizes[4];        // batch has N entries
  const int E = in_sizes[3] / 2;    // edge_index is [2,E]
  const int G = out_size;           // output is [G]

  float* ws = (float*)d_ws;
  const size_t NH = (size_t)N * HD;
  float* h        = ws;
  float* peh      = ws + NH;
  float* aggr     = ws + 2 * NH;
  float* posAggr  = ws + 3 * NH;
  float* pooled   = ws + 4 * NH;

  const int nodeBlocks = (N + TILE - 1) / TILE;
  const int edgeBlocks = (E + TILE - 1) / TILE;
  const int gBlocks    = (G + TILE - 1) / TILE;

  // Param leaf order (JAX sorted-key tree flatten):
  // 5:embedW 6:embedB 7:embed_peW 8:embed_peB
  // layers l=0..3 at 9+16l: msg(W1,b1,W2,b2) pos(...) upd(...) upd_pe(...)
  // 73..76: pre_readout; 77..80: readout
  embed_kernel<<<nodeBlocks, 256, 0, stream>>>(x, pe, F(5), F(6), F(7), F(8), h, peh, N);

  for (int l = 0; l < 4; ++l) {
    int b = 9 + l * 16;
    hipMemsetAsync(aggr, 0, NH * sizeof(float), stream);
    hipMemsetAsync(posAggr, 0, NH * sizeof(float), stream);
    lspe_edge_kernel<<<edgeBlocks, 256, 0, stream>>>(
        h, peh, pos, ei,
        F(b + 0), F(b + 1), F(b + 2), F(b + 3),
        F(b + 4), F(b + 5), F(b + 6), F(b + 7),
        aggr, posAggr, E);
    lspe_node_kernel<<<nodeBlocks, 256, 0, stream>>>(
        h, peh, aggr, posAggr,
        F(b + 8), F(b + 9), F(b + 10), F(b + 11),
        F(b + 12), F(b + 13), F(b + 14), F(b + 15), N);
  }

  hipMemsetAsync(pooled, 0, (size_t)G * HD * sizeof(float), stream);
  prepool_kernel<<<nodeBlocks, 256, 0, stream>>>(h, batch, F(73), F(74), F(75), F(76), pooled, N);
  readout_kernel<<<gBlocks, 256, 0, stream>>>(pooled, F(77), F(78), F(79), F(80), (float*)d_out, G);
}